// VectorQuantizer_76587856823005
// MI455X (gfx1250) — compile-verified
//
#include <hip/hip_runtime.h>
#include <hip/hip_bf16.h>

// Problem constants
#define N_ROWS 65536   // B*H*W = 64*32*32
#define DDIM   64
#define KCB    1024

typedef __attribute__((ext_vector_type(2))) float v2f;
typedef __attribute__((ext_vector_type(8))) float v8f;

// ---------------------------------------------------------------------------
// Kernel 0: prep — zero accumulators, transpose codebook [D,K]->[K,D], cnorm
// ---------------------------------------------------------------------------
__global__ __launch_bounds__(256) void vq_prep_kernel(
    const float* __restrict__ cb,     // [D,K] row-major
    float* __restrict__ cbT,          // [K,D]
    float* __restrict__ cnorm,        // [K]
    unsigned* __restrict__ hist,      // [K]
    float* __restrict__ sqsum)        // [1]
{
    const int gid = blockIdx.x * 256 + threadIdx.x;   // 0..65535
    // transpose (coalesced read, scattered write; one-time 256KB)
    const int d = gid >> 10;
    const int k = gid & (KCB - 1);
    cbT[(size_t)k * DDIM + d] = cb[gid];

    if (gid < KCB) {
        hist[gid] = 0u;
        float s = 0.0f;
        #pragma unroll 4
        for (int dd = 0; dd < DDIM; ++dd) {
            float e = cb[(size_t)dd * KCB + gid];   // coalesced across gid
            s += e * e;
        }
        cnorm[gid] = s;
    }
    if (gid == 0) sqsum[0] = 0.0f;
}

// ---------------------------------------------------------------------------
// Kernel 1: argmin over K via V_WMMA_F32_16X16X4_F32.
// Each wave: 32 rows (two 16-row A tiles resident in VGPRs), scans all 1024
// columns in 16-col tiles. dist = cnorm[k] - 2 * dot(x, e_k)  (|x|^2 dropped).
// Column tiles are double-buffered: tile ct+1's 16 B fragments stream into a
// second register buffer while tile ct's 32 WMMAs execute.
// ---------------------------------------------------------------------------
__global__ __launch_bounds__(128) void vq_argmin_kernel(
    const float* __restrict__ x,      // [N,64]
    const float* __restrict__ cbT,    // [K,64]
    const float* __restrict__ cnorm,  // [K]
    int* __restrict__ idx_out)        // [N]
{
    const int lane    = threadIdx.x & 31;
    const int wave    = blockIdx.x * 4 + (threadIdx.x >> 5);  // 2048 waves
    const int rowbase = wave * 32;
    const int halfsel = lane >> 4;     // 0: K={0,1}, 1: K={2,3} per WMMA step
    const int lrow    = lane & 15;     // M (for A) / N (for B) index

    // --- load both 16x64 A tiles into registers (ISA 16x4 f32 A layout) ---
    v2f a[2][16];
    #pragma unroll
    for (int t = 0; t < 2; ++t) {
        const float* rp = x + (size_t)(rowbase + t * 16 + lrow) * DDIM + 2 * halfsel;
        #pragma unroll
        for (int s = 0; s < 16; ++s)
            a[t][s] = *(const v2f*)(rp + s * 4);
    }

    float bestd[2][8];
    int   besti[2][8];
    #pragma unroll
    for (int t = 0; t < 2; ++t)
        #pragma unroll
        for (int v = 0; v < 8; ++v) { bestd[t][v] = 3.0e38f; besti[t][v] = 0; }

    const v8f vzero = {0.f, 0.f, 0.f, 0.f, 0.f, 0.f, 0.f, 0.f};

    // per-lane base of the B stream: this lane's column within any tile
    const float* bbase = cbT + (size_t)lrow * DDIM + 2 * halfsel;

    auto load_tile = [&](v2f (&b)[16], int ct) {
        const float* p = bbase + (size_t)ct * (16 * DDIM);
        #pragma unroll
        for (int s = 0; s < 16; ++s)
            b[s] = *(const v2f*)(p + s * 4);
    };

    auto compute_tile = [&](const v2f (&b)[16], int ct, float cn) {
        v8f acc0 = vzero, acc1 = vzero;
        #pragma unroll
        for (int s = 0; s < 16; ++s) {               // K = 64 in steps of 4
            acc0 = __builtin_amdgcn_wmma_f32_16x16x4_f32(
                false, a[0][s], false, b[s], (short)0, acc0, false, false);
            acc1 = __builtin_amdgcn_wmma_f32_16x16x4_f32(
                false, a[1][s], false, b[s], (short)0, acc1, false, false);
        }
        const int col = ct * 16 + lrow;
        #pragma unroll
        for (int v = 0; v < 8; ++v) {
            float d0 = cn - 2.0f * acc0[v];
            if (d0 < bestd[0][v]) { bestd[0][v] = d0; besti[0][v] = col; }
            float d1 = cn - 2.0f * acc1[v];
            if (d1 < bestd[1][v]) { bestd[1][v] = d1; besti[1][v] = col; }
        }
    };

    v2f b0[16], b1[16];
    load_tile(b0, 0);
    float cn0 = cnorm[lrow];
    float cn1;

    for (int ct = 0; ct < KCB / 16; ct += 2) {       // 64 tiles, unroll-by-2
        const int n1 = (ct + 1) & 63;
        const int n2 = (ct + 2) & 63;
        load_tile(b1, n1);                           // prefetch ct+1
        cn1 = cnorm[n1 * 16 + lrow];
        compute_tile(b0, ct, cn0);                   // consume ct
        load_tile(b0, n2);                           // prefetch ct+2
        cn0 = cnorm[n2 * 16 + lrow];
        compute_tile(b1, ct + 1, cn1);               // consume ct+1
    }

    // cross-lane min within each 16-lane group (first-index tie-break)
    #pragma unroll
    for (int off = 1; off < 16; off <<= 1) {
        #pragma unroll
        for (int t = 0; t < 2; ++t)
            #pragma unroll
            for (int v = 0; v < 8; ++v) {
                float od = __shfl_xor(bestd[t][v], off, 32);
                int   oi = __shfl_xor(besti[t][v], off, 32);
                if (od < bestd[t][v] ||
                    (od == bestd[t][v] && oi < besti[t][v])) {
                    bestd[t][v] = od; besti[t][v] = oi;
                }
            }
    }

    if (lrow == 0) {
        #pragma unroll
        for (int t = 0; t < 2; ++t)
            #pragma unroll
            for (int v = 0; v < 8; ++v)
                idx_out[rowbase + t * 16 + 8 * halfsel + v] = besti[t][v];
    }
}

// ---------------------------------------------------------------------------
// Kernel 2: gather + ste + squared-error + histogram. One wave per row
// (32 consecutive rows/wave); lane handles 2 elements (float2).
// ---------------------------------------------------------------------------
__global__ __launch_bounds__(256) void vq_gather_kernel(
    const float* __restrict__ x,
    const float* __restrict__ cbT,
    const int* __restrict__ idx,
    float* __restrict__ ste,          // [N,64]
    float* __restrict__ sqsum,        // [1]
    unsigned* __restrict__ hist)      // [K]
{
    const int lane = threadIdx.x & 31;
    const int wave = blockIdx.x * 8 + (threadIdx.x >> 5);   // 2048 waves

    float lsum = 0.0f;
    for (int r = 0; r < 32; ++r) {
        const int row = wave * 32 + r;
        const int k = idx[row];
        v2f q  = *(const v2f*)(cbT + (size_t)k * DDIM + lane * 2);
        v2f xv = *(const v2f*)(x + (size_t)row * DDIM + lane * 2);
        v2f dq; dq.x = q.x - xv.x; dq.y = q.y - xv.y;       // quantized - x
        v2f o;  o.x = xv.x + dq.x; o.y = xv.y + dq.y;       // ste literally
        *(v2f*)(ste + (size_t)row * DDIM + lane * 2) = o;
        lsum += dq.x * dq.x + dq.y * dq.y;
        if (lane == 0) atomicAdd(&hist[k], 1u);
    }
    // wave reduction
    #pragma unroll
    for (int off = 16; off > 0; off >>= 1)
        lsum += __shfl_xor(lsum, off, 32);
    if (lane == 0) atomicAdd(sqsum, lsum);
}

// ---------------------------------------------------------------------------
// Kernel 3: finalize — perplexity + losses into d_out tail
// ---------------------------------------------------------------------------
__global__ __launch_bounds__(1024) void vq_final_kernel(
    const unsigned* __restrict__ hist,
    const float* __restrict__ sqsum,
    float* __restrict__ out3)         // perplexity, codebook_loss, commitment
{
    __shared__ float red[1024];
    const int tid = threadIdx.x;
    const float p = (float)hist[tid] * (1.0f / (float)N_ROWS);
    red[tid] = p * logf(p + 1e-10f);
    __syncthreads();
    for (int s = 512; s > 0; s >>= 1) {
        if (tid < s) red[tid] += red[tid + s];
        __syncthreads();
    }
    if (tid == 0) {
        out3[0] = expf(-red[0]);                                  // perplexity
        const float loss = sqsum[0] * (1.0f / ((float)N_ROWS * (float)DDIM));
        out3[1] = loss;                                           // codebook
        out3[2] = 0.25f * loss;                                   // commitment
    }
}

// ---------------------------------------------------------------------------
extern "C" void kernel_launch(void* const* d_in, const int* in_sizes, int n_in,
                              void* d_out, int out_size, void* d_ws, size_t ws_size,
                              hipStream_t stream) {
    (void)in_sizes; (void)n_in; (void)out_size; (void)ws_size;
    const float* x  = (const float*)d_in[0];   // [64,32,32,64] f32
    const float* cb = (const float*)d_in[1];   // [64,1024] f32
    float* out = (float*)d_out;                // ste .. then 3 scalars

    // workspace layout (floats): sqsum(1) pad->4 | hist 1024 | cnorm 1024 |
    // cbT 65536 | idx 65536   (~520 KB total)
    float*    ws    = (float*)d_ws;
    float*    sqsum = ws;
    unsigned* hist  = (unsigned*)(ws + 4);
    float*    cnorm = ws + 4 + KCB;
    float*    cbT   = ws + 4 + 2 * KCB;
    int*      idx   = (int*)(ws + 4 + 2 * KCB + KCB * DDIM);

    vq_prep_kernel<<<(DDIM * KCB) / 256, 256, 0, stream>>>(cb, cbT, cnorm, hist, sqsum);
    vq_argmin_kernel<<<512, 128, 0, stream>>>(x, cbT, cnorm, idx);      // 2048 waves * 32 rows
    vq_gather_kernel<<<256, 256, 0, stream>>>(x, cbT, idx, out, sqsum, hist);
    vq_final_kernel<<<1, 1024, 0, stream>>>(hist, sqsum, out + (size_t)N_ROWS * DDIM);
}